// DGCNN_84250078478829
// MI455X (gfx1250) — compile-verified
//
#include <hip/hip_runtime.h>
#include <hip/hip_bf16.h>

#define B_SZ   8
#define N_PTS  2048
#define K_NN   20
#define CAT_C  512
#define BN_EPS 1e-5f
#define NSLOPE 0.2f

typedef float v2f __attribute__((ext_vector_type(2)));
typedef float v8f __attribute__((ext_vector_type(8)));

// ---------------------------------------------------------------------------
// Transpose x (B,3,N) -> xt (B,N,3)
// ---------------------------------------------------------------------------
__global__ void transpose_x_kernel(const float* __restrict__ x, float* __restrict__ xt) {
    int t = blockIdx.x * blockDim.x + threadIdx.x;
    if (t >= B_SZ * N_PTS) return;
    int b = t / N_PTS, n = t % N_PTS;
    #pragma unroll
    for (int c = 0; c < 3; ++c)
        xt[t * 3 + c] = x[(b * 3 + c) * N_PTS + n];
}

// ---------------------------------------------------------------------------
// Fill / pad helpers
// ---------------------------------------------------------------------------
__global__ void fill_f32_kernel(float* __restrict__ p, float v, int n) {
    int i = blockIdx.x * blockDim.x + threadIdx.x;
    if (i < n) p[i] = v;
}

__global__ void fill_slice_kernel(float* __restrict__ base, int ld, int coff, int C,
                                  float v, int total) {
    int i = blockIdx.x * blockDim.x + threadIdx.x;
    if (i >= total) return;
    int r = i / C, c = i - r * C;
    base[r * ld + coff + c] = v;
}

// W (C_out, C_e) -> Wp (C_out, C_e_pad), zero-padded columns
__global__ void pad_w_kernel(const float* __restrict__ W, float* __restrict__ Wp,
                             int C_e, int C_e_pad, int total) {
    int i = blockIdx.x * blockDim.x + threadIdx.x;
    if (i >= total) return;
    int c = i / C_e_pad, k = i - c * C_e_pad;
    Wp[i] = (k < C_e) ? W[c * C_e + k] : 0.f;
}

// ---------------------------------------------------------------------------
// KNN: one thread per query point, LDS-tiled candidates, top-20 insertion sort
// ---------------------------------------------------------------------------
template <int C>
__global__ void knn_kernel(const float* __restrict__ fin, int ldf, int* __restrict__ out_idx) {
    constexpr int QT = 64;
    constexpr int CT = 32;
    __shared__ float qs[QT * C];
    __shared__ float cs[CT * C];

    int blocksPerBatch = N_PTS / QT;
    int b  = blockIdx.x / blocksPerBatch;
    int q0 = (blockIdx.x % blocksPerBatch) * QT;
    int t  = threadIdx.x;

    for (int i = t; i < QT * C; i += QT) {
        int qq = i / C, c = i - qq * C;
        qs[i] = fin[(b * N_PTS + q0 + qq) * ldf + c];
    }
    __syncthreads();

    float bd[K_NN];
    int   bi[K_NN];
    #pragma unroll
    for (int j = 0; j < K_NN; ++j) { bd[j] = 3.4e38f; bi[j] = 0; }

    for (int m0 = 0; m0 < N_PTS; m0 += CT) {
        for (int i = t; i < CT * C; i += QT) {
            int mm = i / C, c = i - mm * C;
            cs[i] = fin[(b * N_PTS + m0 + mm) * ldf + c];
        }
        __syncthreads();
        for (int mm = 0; mm < CT; ++mm) {
            float d = 0.f;
            #pragma unroll
            for (int c = 0; c < C; ++c) {
                float df = qs[t * C + c] - cs[mm * C + c];
                d += df * df;
            }
            if (d < bd[K_NN - 1]) {
                int j = K_NN - 1;
                while (j > 0 && bd[j - 1] > d) {
                    bd[j] = bd[j - 1]; bi[j] = bi[j - 1]; --j;
                }
                bd[j] = d; bi[j] = m0 + mm;
            }
        }
        __syncthreads();
    }

    int n = q0 + t;
    #pragma unroll
    for (int j = 0; j < K_NN; ++j)
        out_idx[(b * N_PTS + n) * K_NN + j] = bi[j];
}

// ---------------------------------------------------------------------------
// Branch-free WMMA k-loop: A rows from LDS (v2f ds_load), B cols from padded
// global weights (v2f global_load).  C_e_pad % 4 == 0, everything 8B aligned.
// ---------------------------------------------------------------------------
__device__ __forceinline__ v8f wmma_loop(const float* __restrict__ erow,
                                         const float* __restrict__ wrow,
                                         int C_e_pad) {
    v8f acc = {};
    #pragma unroll 4
    for (int ks0 = 0; ks0 < C_e_pad; ks0 += 4) {
        v2f a  = *(const v2f*)(erow + ks0);
        v2f bv = *(const v2f*)(wrow + ks0);
        acc = __builtin_amdgcn_wmma_f32_16x16x4_f32(false, a, false, bv,
                                                    (short)0, acc, false, false);
    }
    return acc;
}

__device__ __forceinline__ void stats_reduce(v8f acc, int lane, int c0,
                                             float* __restrict__ gsum,
                                             float* __restrict__ gsumsq) {
    float s = 0.f, s2 = 0.f;
    #pragma unroll
    for (int v = 0; v < 8; ++v) { float y = acc[v]; s += y; s2 += y * y; }
    s  += __shfl_xor(s, 16, 32);
    s2 += __shfl_xor(s2, 16, 32);
    if (lane < 16) {
        atomicAdd(&gsum[c0 + lane], s);
        atomicAdd(&gsumsq[c0 + lane], s2);
    }
}

// ---------------------------------------------------------------------------
// Edge GEMM tiles: block = 128 threads (4 waves), one 16-row E tile in LDS,
// 4 adjacent 16-wide column tiles (64 output channels) per block.
// ---------------------------------------------------------------------------
__device__ __forceinline__ void build_edge_tile(float* __restrict__ elds,
                                                int* __restrict__ sBaseN,
                                                int* __restrict__ sBaseNbr,
                                                const float* __restrict__ fin, int ldf,
                                                const int* __restrict__ nbr_idx,
                                                int C_e, int C_e_pad, int b, int r0) {
    int t  = threadIdx.x;
    int ch = C_e >> 1;
    if (t < 16) {
        int row = r0 + t;
        int n   = row / K_NN;
        int kk  = row - n * K_NN;
        int nb  = nbr_idx[(b * N_PTS + n) * K_NN + kk];
        sBaseN[t]   = (b * N_PTS + n)  * ldf;
        sBaseNbr[t] = (b * N_PTS + nb) * ldf;
    }
    __syncthreads();
    for (int i = t; i < 16 * C_e_pad; i += 128) {
        int r = i / C_e_pad, c = i - r * C_e_pad;
        float v = 0.f;
        if (c < ch)          v = fin[sBaseNbr[r] + c] - fin[sBaseN[r] + c];
        else if (c < C_e)    v = fin[sBaseN[r] + (c - ch)];
        elds[i] = v;
    }
    __syncthreads();
}

__global__ void edge_gemm_stats_kernel(const float* __restrict__ fin, int ldf,
                                       const int* __restrict__ nbr_idx,
                                       const float* __restrict__ Wp,
                                       int C_e, int C_e_pad, int colGroups,
                                       float* __restrict__ gsum,
                                       float* __restrict__ gsumsq) {
    __shared__ float elds[16 * 256];
    __shared__ int   sBaseN[16], sBaseNbr[16];
    int b  = blockIdx.y;
    int cg = blockIdx.x % colGroups;
    int rt = blockIdx.x / colGroups;
    int r0 = rt * 16;

    build_edge_tile(elds, sBaseN, sBaseNbr, fin, ldf, nbr_idx, C_e, C_e_pad, b, r0);

    int lane = threadIdx.x & 31;
    int wv   = threadIdx.x >> 5;
    int c0   = (cg * 4 + wv) * 16;
    int half = lane >> 4, lidx = lane & 15;
    int ko   = 2 * half;

    v8f acc = wmma_loop(elds + lidx * C_e_pad + ko,
                        Wp + (c0 + lidx) * C_e_pad + ko, C_e_pad);
    stats_reduce(acc, lane, c0, gsum, gsumsq);
}

__global__ void edge_gemm_apply_kernel(const float* __restrict__ fin, int ldf,
                                       const int* __restrict__ nbr_idx,
                                       const float* __restrict__ Wp,
                                       int C_e, int C_e_pad, int colGroups,
                                       const float* __restrict__ scale,
                                       const float* __restrict__ shift,
                                       float* __restrict__ outSlice, int ldo) {
    __shared__ float elds[16 * 256];
    __shared__ int   sBaseN[16], sBaseNbr[16];
    int b  = blockIdx.y;
    int cg = blockIdx.x % colGroups;
    int rt = blockIdx.x / colGroups;
    int r0 = rt * 16;

    build_edge_tile(elds, sBaseN, sBaseNbr, fin, ldf, nbr_idx, C_e, C_e_pad, b, r0);

    int lane = threadIdx.x & 31;
    int wv   = threadIdx.x >> 5;
    int c0   = (cg * 4 + wv) * 16;
    int half = lane >> 4, lidx = lane & 15;
    int ko   = 2 * half;

    v8f acc = wmma_loop(elds + lidx * C_e_pad + ko,
                        Wp + (c0 + lidx) * C_e_pad + ko, C_e_pad);

    int   c  = c0 + lidx;
    float sc = scale[c], sh = shift[c];
    #pragma unroll
    for (int v = 0; v < 8; ++v) {
        int   row = r0 + 8 * half + v;     // C/D layout: lanes>=16 hold rows 8..15
        int   n   = row / K_NN;
        float y   = acc[v] * sc + sh;
        y = (y > 0.f) ? y : NSLOPE * y;
        atomicMax(&outSlice[(b * N_PTS + n) * ldo + c], y);
    }
}

// ---------------------------------------------------------------------------
// Final flat GEMM: 16 x 512 row tile in LDS shared by 4 waves (4 col tiles).
// ---------------------------------------------------------------------------
__device__ __forceinline__ void build_flat_tile(float* __restrict__ elds,
                                                const float* __restrict__ fin, int ldf,
                                                int b, int r0) {
    int t = threadIdx.x;
    for (int i = t; i < 16 * CAT_C; i += 128) {
        int r = i / CAT_C, c = i - r * CAT_C;
        elds[i] = fin[(b * N_PTS + r0 + r) * ldf + c];
    }
    __syncthreads();
}

__global__ void flat_gemm_stats_kernel(const float* __restrict__ fin, int ldf,
                                       const float* __restrict__ W,
                                       int colGroups,
                                       float* __restrict__ gsum,
                                       float* __restrict__ gsumsq) {
    __shared__ float elds[16 * CAT_C];
    int b  = blockIdx.y;
    int cg = blockIdx.x % colGroups;
    int rt = blockIdx.x / colGroups;
    int r0 = rt * 16;

    build_flat_tile(elds, fin, ldf, b, r0);

    int lane = threadIdx.x & 31;
    int wv   = threadIdx.x >> 5;
    int c0   = (cg * 4 + wv) * 16;
    int half = lane >> 4, lidx = lane & 15;
    int ko   = 2 * half;

    v8f acc = wmma_loop(elds + lidx * CAT_C + ko,
                        W + (c0 + lidx) * CAT_C + ko, CAT_C);
    stats_reduce(acc, lane, c0, gsum, gsumsq);
}

__global__ void flat_gemm_apply_kernel(const float* __restrict__ fin, int ldf,
                                       const float* __restrict__ W,
                                       int C_out, int colGroups,
                                       const float* __restrict__ scale,
                                       const float* __restrict__ shift,
                                       float* __restrict__ feat,
                                       float* __restrict__ pooled) {
    __shared__ float elds[16 * CAT_C];
    int b  = blockIdx.y;
    int cg = blockIdx.x % colGroups;
    int rt = blockIdx.x / colGroups;
    int r0 = rt * 16;

    build_flat_tile(elds, fin, ldf, b, r0);

    int lane = threadIdx.x & 31;
    int wv   = threadIdx.x >> 5;
    int c0   = (cg * 4 + wv) * 16;
    int half = lane >> 4, lidx = lane & 15;
    int ko   = 2 * half;

    v8f acc = wmma_loop(elds + lidx * CAT_C + ko,
                        W + (c0 + lidx) * CAT_C + ko, CAT_C);

    int   c  = c0 + lidx;
    float sc = scale[c], sh = shift[c];
    float m  = -3.4e38f;
    #pragma unroll
    for (int v = 0; v < 8; ++v) {
        int   n = r0 + 8 * half + v;
        float y = acc[v] * sc + sh;
        y = (y > 0.f) ? y : NSLOPE * y;
        feat[(b * C_out + c) * N_PTS + n] = y;
        m = fmaxf(m, y);
    }
    atomicMax(&pooled[b * C_out + c], m);
}

// ---------------------------------------------------------------------------
// BN scale/shift from accumulated stats
// ---------------------------------------------------------------------------
__global__ void finalize_bn_kernel(const float* __restrict__ gsum,
                                   const float* __restrict__ gsumsq,
                                   const float* __restrict__ gamma,
                                   const float* __restrict__ beta,
                                   float* __restrict__ scale, float* __restrict__ shift,
                                   int C_out, float invCount) {
    int c = blockIdx.x * blockDim.x + threadIdx.x;
    if (c >= C_out) return;
    float mean = gsum[c] * invCount;
    float var  = gsumsq[c] * invCount - mean * mean;
    float sc   = gamma[c] * rsqrtf(var + BN_EPS);
    scale[c] = sc;
    shift[c] = beta[c] - mean * sc;
}

// ---------------------------------------------------------------------------
// Host orchestration
// ---------------------------------------------------------------------------
extern "C" void kernel_launch(void* const* d_in, const int* in_sizes, int n_in,
                              void* d_out, int out_size, void* d_ws, size_t ws_size,
                              hipStream_t stream) {
    (void)in_sizes; (void)n_in; (void)out_size; (void)ws_size;

    const float* x  = (const float*)d_in[0];
    const float* w1 = (const float*)d_in[1];
    const float* g1 = (const float*)d_in[2];
    const float* b1 = (const float*)d_in[3];
    const float* w2 = (const float*)d_in[4];
    const float* g2 = (const float*)d_in[5];
    const float* b2 = (const float*)d_in[6];
    const float* w3 = (const float*)d_in[7];
    const float* g3 = (const float*)d_in[8];
    const float* b3 = (const float*)d_in[9];
    const float* w4 = (const float*)d_in[10];
    const float* g4 = (const float*)d_in[11];
    const float* b4 = (const float*)d_in[12];
    const float* w5 = (const float*)d_in[13];
    const float* g5 = (const float*)d_in[14];
    const float* b5 = (const float*)d_in[15];

    // workspace layout (floats)
    float* ws     = (float*)d_ws;
    float* xt     = ws;                                         // B*N*3
    float* cat    = xt + (size_t)B_SZ * N_PTS * 3;              // B*N*512
    int*   idxb   = (int*)(cat + (size_t)B_SZ * N_PTS * CAT_C); // B*N*20
    float* gsum   = (float*)(idxb + (size_t)B_SZ * N_PTS * K_NN);
    float* gsumsq = gsum + 1024;
    float* scale  = gsumsq + 1024;
    float* shiftb = scale + 1024;
    float* wpad   = shiftb + 1024;                              // up to 256*256

    float* pooled = (float*)d_out;
    float* feat   = pooled + (size_t)B_SZ * 1024;

    const int ROWS_E = N_PTS * K_NN;   // 40960 edge rows per batch
    const int RT_E   = ROWS_E / 16;    // 2560
    const int RT_F   = N_PTS / 16;     // 128
    const float invE = 1.f / (float)(B_SZ * N_PTS * K_NN);
    const float invF = 1.f / (float)(B_SZ * N_PTS);
    const int knnBlocks = B_SZ * (N_PTS / 64);

    transpose_x_kernel<<<(B_SZ * N_PTS + 255) / 256, 256, 0, stream>>>(x, xt);

    struct Stage {
        const float* fin; int ldf; int C_in; int C_e; int C_e_pad; int C_out;
        const float* W; const float* g; const float* bta; int coff;
    };
    const Stage st[4] = {
        { xt,        3,   3,   6,   8,   64,  w1, g1, b1, 0   },
        { cat + 0,   512, 64,  128, 128, 64,  w2, g2, b2, 64  },
        { cat + 64,  512, 64,  128, 128, 128, w3, g3, b3, 128 },
        { cat + 128, 512, 128, 256, 256, 256, w4, g4, b4, 256 },
    };

    for (int s = 0; s < 4; ++s) {
        const Stage& S = st[s];

        // 1) KNN on this stage's input features
        if (S.C_in == 3)
            knn_kernel<3><<<knnBlocks, 64, 0, stream>>>(S.fin, S.ldf, idxb);
        else if (S.C_in == 64)
            knn_kernel<64><<<knnBlocks, 64, 0, stream>>>(S.fin, S.ldf, idxb);
        else
            knn_kernel<128><<<knnBlocks, 64, 0, stream>>>(S.fin, S.ldf, idxb);

        // 2) pad weights, zero BN accumulators, init output slice for max
        int wTot = S.C_out * S.C_e_pad;
        pad_w_kernel<<<(wTot + 255) / 256, 256, 0, stream>>>(S.W, wpad, S.C_e, S.C_e_pad, wTot);
        fill_f32_kernel<<<(2048 + 255) / 256, 256, 0, stream>>>(gsum, 0.f, 2048);
        int totOut = B_SZ * N_PTS * S.C_out;
        fill_slice_kernel<<<(totOut + 255) / 256, 256, 0, stream>>>(
            cat, CAT_C, S.coff, S.C_out, -3.4e38f, totOut);

        // 3) stats pass (WMMA)
        int colGroups = S.C_out / 64;
        dim3 grid(colGroups * RT_E, B_SZ);
        edge_gemm_stats_kernel<<<grid, 128, 0, stream>>>(
            S.fin, S.ldf, idxb, wpad, S.C_e, S.C_e_pad, colGroups, gsum, gsumsq);

        // 4) scale/shift
        finalize_bn_kernel<<<(S.C_out + 63) / 64, 64, 0, stream>>>(
            gsum, gsumsq, S.g, S.bta, scale, shiftb, S.C_out, invE);

        // 5) apply pass (WMMA) + max over K
        edge_gemm_apply_kernel<<<grid, 128, 0, stream>>>(
            S.fin, S.ldf, idxb, wpad, S.C_e, S.C_e_pad, colGroups, scale, shiftb,
            cat + S.coff, CAT_C);
    }

    // ---- final: y = cat * w5^T, BN over (B,N), lrelu, feat + pooled ----
    fill_f32_kernel<<<(2048 + 255) / 256, 256, 0, stream>>>(gsum, 0.f, 2048);
    {
        int colGroups = 1024 / 64;   // 16
        dim3 grid(colGroups * RT_F, B_SZ);
        flat_gemm_stats_kernel<<<grid, 128, 0, stream>>>(
            cat, CAT_C, w5, colGroups, gsum, gsumsq);

        finalize_bn_kernel<<<(1024 + 63) / 64, 64, 0, stream>>>(
            gsum, gsumsq, g5, b5, scale, shiftb, 1024, invF);

        fill_f32_kernel<<<(B_SZ * 1024 + 255) / 256, 256, 0, stream>>>(
            pooled, -3.4e38f, B_SZ * 1024);

        flat_gemm_apply_kernel<<<grid, 128, 0, stream>>>(
            cat, CAT_C, w5, 1024, colGroups, scale, shiftb, feat, pooled);
    }
}